// EHCF_86062554677780
// MI455X (gfx1250) — compile-verified
//
#include <hip/hip_runtime.h>
#include <hip/hip_bf16.h>

typedef __attribute__((ext_vector_type(2))) float v2f;
typedef __attribute__((ext_vector_type(8))) float v8f;

#define NUM_B   1024
#define NUM_N   50000L
#define DDIM    64

// ---------------------------------------------------------------------------
// Kernel 1: cascade the behavior weight vectors -> W3[3][64] in workspace.
// ---------------------------------------------------------------------------
__global__ __launch_bounds__(64) void prep_weights(
    const float* __restrict__ wv,   // weight_view [64]
    const float* __restrict__ Wvc,  // [64,64]
    const float* __restrict__ Wvb,  // [64,64]
    const float* __restrict__ Wcb,  // [64,64]
    const float* __restrict__ bvc,  // [64]
    const float* __restrict__ bvb,  // [64]
    const float* __restrict__ bcb,  // [64]
    float* __restrict__ W3)         // out [3][64]
{
    __shared__ float s_wv[DDIM];
    __shared__ float s_wc[DDIM];
    const int d = threadIdx.x;  // 0..63

    s_wv[d] = wv[d];
    __syncthreads();

    // w_cart = W_vc @ w_view + b_vc
    float wc = bvc[d];
    #pragma unroll 8
    for (int j = 0; j < DDIM; ++j) wc += Wvc[d * DDIM + j] * s_wv[j];
    s_wc[d] = wc;
    __syncthreads();

    // w_buy = W_vb @ w_view + b_vb + W_cb @ w_cart + b_cb
    float wb = bvb[d] + bcb[d];
    #pragma unroll 8
    for (int j = 0; j < DDIM; ++j)
        wb += Wvb[d * DDIM + j] * s_wv[j] + Wcb[d * DDIM + j] * s_wc[j];

    W3[0 * DDIM + d] = s_wv[d];
    W3[1 * DDIM + d] = wc;
    W3[2 * DDIM + d] = wb;
}

// ---------------------------------------------------------------------------
// Kernel 2: A[k][b][d] = emb_user[batch_users[b], d] * drop_mask[b,d] * W3[k,d]
// A is [3][1024][64] f32 in workspace (0.75 MB, lives in L2 for the GEMM).
// ---------------------------------------------------------------------------
__global__ __launch_bounds__(256) void prep_A(
    const int*   __restrict__ users,     // [1024]
    const float* __restrict__ emb_user,  // [1e6, 64]
    const float* __restrict__ drop,      // [1024, 64]
    const float* __restrict__ W3,        // [3, 64]
    float*       __restrict__ A)         // [3, 1024, 64]
{
    const int idx = blockIdx.x * 256 + threadIdx.x;   // b*64 + d, 65536 total
    const int b = idx >> 6;
    const int d = idx & 63;
    const long urow = (long)users[b] * DDIM + d;
    const float u = emb_user[urow] * drop[idx];
    A[idx]                     = u * W3[d];
    A[NUM_B * DDIM + idx]      = u * W3[DDIM + d];
    A[2 * NUM_B * DDIM + idx]  = u * W3[2 * DDIM + d];
}

// ---------------------------------------------------------------------------
// Kernel 3: lik[k, b0:b0+16, n0:n0+16] = A_k-tile @ v-tile^T via fp32 WMMA.
// One wave per 16x16 tile; 3 accumulators (k=view/cart/buy) share B fragments.
// blockDim = (32, 8): 8 waves/WG cover 8 consecutive b-tiles of one n-tile,
// so the item tile streams once through the cache per workgroup.
// ---------------------------------------------------------------------------
__global__ __launch_bounds__(256) void ehcf_gemm(
    const float* __restrict__ A,           // [3, 1024, 64]
    const float* __restrict__ emb_item,    // [50000, 64]
    const int*   __restrict__ whole_items, // [50000]
    float*       __restrict__ out)         // [3, 1024, 50000]
{
    const int lane   = threadIdx.x;                 // 0..31
    const int n_tile = blockIdx.x;                  // 0..3124
    const int b_tile = blockIdx.y * 8 + threadIdx.y;// 0..63
    const int n0 = n_tile * 16;
    const int b0 = b_tile * 16;

    // A 16x4 fp32 fragment layout: lane L -> M = L&15, K = (L>>4)*2 + {0,1}
    // B 4x16 fp32 fragment layout: lane L -> N = L&15, K = (L>>4)*2 + {0,1}
    const int mrow = lane & 15;
    const int koff = (lane >> 4) * 2;

    const long item = (long)whole_items[n0 + mrow];
    const float* vrow = emb_item + item * DDIM + koff;

    const float* a0 = A + (long)(b0 + mrow) * DDIM + koff;
    const float* a1 = a0 + (long)NUM_B * DDIM;
    const float* a2 = a1 + (long)NUM_B * DDIM;

    v8f c0 = {}, c1 = {}, c2 = {};

    #pragma unroll
    for (int t = 0; t < 16; ++t) {                  // K = 64 = 16 steps of 4
        const v2f bf  = *(const v2f*)(vrow + 4 * t);
        const v2f af0 = *(const v2f*)(a0   + 4 * t);
        const v2f af1 = *(const v2f*)(a1   + 4 * t);
        const v2f af2 = *(const v2f*)(a2   + 4 * t);
        // (neg_a, A, neg_b, B, c_mod, C, reuse_a, reuse_b)
        c0 = __builtin_amdgcn_wmma_f32_16x16x4_f32(false, af0, false, bf,
                                                   (short)0, c0, false, false);
        c1 = __builtin_amdgcn_wmma_f32_16x16x4_f32(false, af1, false, bf,
                                                   (short)0, c1, false, false);
        c2 = __builtin_amdgcn_wmma_f32_16x16x4_f32(false, af2, false, bf,
                                                   (short)0, c2, false, false);
    }

    // C/D layout: lane L, VGPR r -> M = r + 8*(L>>4), N = L&15
    const int  col   = lane & 15;
    const int  rbase = (lane >> 4) * 8;
    const long BN    = (long)NUM_B * NUM_N;
    float* o = out + (long)(b0 + rbase) * NUM_N + (n0 + col);

    #pragma unroll
    for (int r = 0; r < 8; ++r) {
        o[(long)r * NUM_N]            = c0[r];
        o[BN + (long)r * NUM_N]       = c1[r];
        o[2 * BN + (long)r * NUM_N]   = c2[r];
    }
}

// ---------------------------------------------------------------------------
extern "C" void kernel_launch(void* const* d_in, const int* in_sizes, int n_in,
                              void* d_out, int out_size, void* d_ws, size_t ws_size,
                              hipStream_t stream) {
    const int*   batch_users = (const int*)  d_in[0];
    const int*   whole_items = (const int*)  d_in[1];
    const float* emb_user    = (const float*)d_in[2];
    const float* emb_item    = (const float*)d_in[3];
    const float* weight_view = (const float*)d_in[4];
    const float* W_vc        = (const float*)d_in[5];
    const float* W_vb        = (const float*)d_in[6];
    const float* W_cb        = (const float*)d_in[7];
    const float* b_vc        = (const float*)d_in[8];
    const float* b_vb        = (const float*)d_in[9];
    const float* b_cb        = (const float*)d_in[10];
    const float* drop_mask   = (const float*)d_in[11];
    float* out = (float*)d_out;

    float* W3 = (float*)d_ws;        // 3*64 floats, padded to 256
    float* A  = W3 + 256;            // 3*1024*64 floats (~0.75 MB)

    prep_weights<<<1, 64, 0, stream>>>(weight_view, W_vc, W_vb, W_cb,
                                       b_vc, b_vb, b_cb, W3);

    prep_A<<<(NUM_B * DDIM) / 256, 256, 0, stream>>>(batch_users, emb_user,
                                                     drop_mask, W3, A);

    dim3 grid(3125, 8);   // 3125 n-tiles x (64 b-tiles / 8 per WG)
    dim3 block(32, 8);    // 8 wave32s
    ehcf_gemm<<<grid, block, 0, stream>>>(A, emb_item, whole_items, out);
}